// NearestNeighborSampler_35012573397448
// MI455X (gfx1250) — compile-verified
//
#include <hip/hip_runtime.h>
#include <cstddef>

typedef __attribute__((ext_vector_type(2))) float v2f;
typedef __attribute__((ext_vector_type(8))) float v8f;

#define Q_MAX  32768
#define B_ROWS 1024
#define DIM    256
#define KNN    15
#define NTILES (Q_MAX / 16)   // 2048
#define NWAVES 8
#define RSTRIDE 36            // dwords per scan row (2 tiles = 32 floats + pad)

// -------- Kernel 1: squared norms of all data_queue rows into d_ws --------
__global__ __launch_bounds__(256) void nns_norms_kernel(
    const float* __restrict__ data, const float* __restrict__ queue,
    float* __restrict__ norms) {
  const int wave = threadIdx.x >> 5;
  const int lane = threadIdx.x & 31;
  const int row  = blockIdx.x * 8 + wave;            // 0..32767
  const float* src = (row < B_ROWS) ? (data + (size_t)row * DIM)
                                    : (queue + (size_t)(row - B_ROWS) * DIM);
  const float4* s4 = (const float4*)src;
  float4 p0 = s4[lane * 2 + 0];
  float4 p1 = s4[lane * 2 + 1];
  float s = p0.x * p0.x + p0.y * p0.y + p0.z * p0.z + p0.w * p0.w
          + p1.x * p1.x + p1.y * p1.y + p1.z * p1.z + p1.w * p1.w;
  s += __shfl_xor(s, 16, 32);
  s += __shfl_xor(s, 8, 32);
  s += __shfl_xor(s, 4, 32);
  s += __shfl_xor(s, 2, 32);
  s += __shfl_xor(s, 1, 32);
  if (lane == 0) norms[row] = s;
}

// -------- Kernel 2: fused WMMA distance GEMM + per-row top-15 + gather --------
__global__ __launch_bounds__(256) void nns_knn_kernel(
    const float* __restrict__ data, const float* __restrict__ queue,
    const int* __restrict__ jdx, const float* __restrict__ norms,
    float* __restrict__ out) {
  __shared__ float tile[NWAVES][16 * RSTRIDE];  // dual-tile scan rows, padded
  __shared__ float topd[NWAVES][16][KNN];
  __shared__ int   topi[NWAVES][16][KNN];
  __shared__ int   qsel[16];

  const int tid   = threadIdx.x;
  const int wave  = tid >> 5;
  const int lane  = tid & 31;
  const int ln    = lane & 15;
  const int hi    = lane >> 4;               // 0: rows 0-7 / K 0-1, 1: rows 8-15 / K 2-3
  const int mbase = blockIdx.x * 16;         // M block (64 blocks)
  const float INF = __builtin_huge_valf();

  // init per-wave top lists (lane r of lower half owns row r)
  if (hi == 0) {
    #pragma unroll
    for (int t = 0; t < KNN; ++t) { topd[wave][ln][t] = INF; topi[wave][ln][t] = 0; }
  }

  // ||x||^2 for the 8 C-fragment rows this lane covers
  float x2v[8];
  #pragma unroll
  for (int i = 0; i < 8; ++i) x2v[i] = norms[mbase + i + 8 * hi];

  const float* arow = data + (size_t)(mbase + ln) * DIM;  // A: M = ln
  const int koff = 2 * hi;                                 // K sub-offset in fragment
  const int mg   = mbase + ln;                             // diagonal column for scan lane

  float curMax = INF;
  int   maxPos = 0;

  // two N-tiles per iteration: independent WMMA chains, shared A fragments
  for (int j0 = wave * 2; j0 < NTILES; j0 += 2 * NWAVES) { // 128 iterations
    const int colbase = j0 * 16;
    const int qcol0 = colbase + ln;
    const int qcol1 = qcol0 + 16;
    const float* brow0 = (qcol0 < B_ROWS) ? (data + (size_t)qcol0 * DIM)
                                          : (queue + (size_t)(qcol0 - B_ROWS) * DIM);
    const float* brow1 = (qcol1 < B_ROWS) ? (data + (size_t)qcol1 * DIM)
                                          : (queue + (size_t)(qcol1 - B_ROWS) * DIM);
    v8f c0 = {0.f, 0.f, 0.f, 0.f, 0.f, 0.f, 0.f, 0.f};
    v8f c1 = {0.f, 0.f, 0.f, 0.f, 0.f, 0.f, 0.f, 0.f};
    #pragma unroll 8
    for (int kk = 0; kk < 64; ++kk) {                      // K = 256, 4 per step
      const int k = kk * 4 + koff;
      v2f a  = *(const v2f*)(arow + k);
      v2f b0 = *(const v2f*)(brow0 + k);
      v2f b1 = *(const v2f*)(brow1 + k);
      c0 = __builtin_amdgcn_wmma_f32_16x16x4_f32(false, a, false, b0,
                                                 (short)0, c0, false, false);
      c1 = __builtin_amdgcn_wmma_f32_16x16x4_f32(false, a, false, b1,
                                                 (short)0, c1, false, false);
    }

    // d^2 = ||x||^2 + ||y||^2 - 2 x.y ; dump both tiles (wave-private buffer)
    const float y20 = norms[qcol0];
    const float y21 = norms[qcol1];
    #pragma unroll
    for (int i = 0; i < 8; ++i) {
      const int r = i + 8 * hi;
      tile[wave][r * RSTRIDE + ln]      = x2v[i] + y20 - 2.0f * c0[i];
      tile[wave][r * RSTRIDE + 16 + ln] = x2v[i] + y21 - 2.0f * c1[i];
    }
    __builtin_amdgcn_wave_barrier();
    __asm__ volatile("" ::: "memory");   // intra-wave LDS is in-order; fence the compiler

    if (hi == 0) {                       // lane r scans row r: 8x b128, one wait
      float rv[32];
      #pragma unroll
      for (int g = 0; g < 8; ++g) {
        const float4 t4 = *(const float4*)&tile[wave][ln * RSTRIDE + g * 4];
        rv[g * 4 + 0] = t4.x; rv[g * 4 + 1] = t4.y;
        rv[g * 4 + 2] = t4.z; rv[g * 4 + 3] = t4.w;
      }
      #pragma unroll
      for (int cc = 0; cc < 32; ++cc) {
        const int col = colbase + cc;
        const float d = rv[cc];
        if (col != mg && d < curMax) {   // rare insert path
          topd[wave][ln][maxPos] = d;
          topi[wave][ln][maxPos] = col;
          float mx = topd[wave][ln][0]; int mp = 0;
          #pragma unroll
          for (int t = 1; t < KNN; ++t) {
            const float v = topd[wave][ln][t];
            if (v > mx) { mx = v; mp = t; }
          }
          curMax = mx; maxPos = mp;
        }
      }
    }
    __asm__ volatile("" ::: "memory");
    __builtin_amdgcn_wave_barrier();
  }

  __syncthreads();

  // merge 8 per-wave lists per row; select rank jdx[row] (tie-break: lower index)
  if (tid < 16) {
    int jr = jdx[mbase + tid];
    if (jr < 0) jr = 0;
    if (jr > KNN - 1) jr = KNN - 1;
    int selCol = 0;
    for (int rank = 0; rank <= jr; ++rank) {
      float bd = INF; int bw = 0, bt = 0, bc = 0x7fffffff;
      for (int w = 0; w < NWAVES; ++w) {
        for (int t = 0; t < KNN; ++t) {
          const float d = topd[w][tid][t];
          const int  ci = topi[w][tid][t];
          if (d < bd || (d == bd && ci < bc)) { bd = d; bc = ci; bw = w; bt = t; }
        }
      }
      selCol = bc;
      topd[bw][tid][bt] = INF;   // remove for next rank
    }
    qsel[tid] = selCol;
  }
  __syncthreads();

  // gather: 16 threads per output row, float4 copies (256 floats per row)
  {
    const int m   = tid >> 4;
    const int sub = tid & 15;
    const int q   = qsel[m];
    const float* src = (q < B_ROWS) ? (data + (size_t)q * DIM)
                                    : (queue + (size_t)(q - B_ROWS) * DIM);
    float* dst = out + (size_t)(mbase + m) * DIM;
    #pragma unroll
    for (int v = sub; v < 64; v += 16) {
      ((float4*)dst)[v] = ((const float4*)src)[v];
    }
  }
}

extern "C" void kernel_launch(void* const* d_in, const int* in_sizes, int n_in,
                              void* d_out, int out_size, void* d_ws, size_t ws_size,
                              hipStream_t stream) {
  const float* data  = (const float*)d_in[0];   // [1024, 256] f32
  const float* queue = (const float*)d_in[1];   // [32768, 256] f32
  const int*   jdx   = (const int*)d_in[2];     // [1024] i32
  float* out   = (float*)d_out;                 // [1024, 256] f32
  float* norms = (float*)d_ws;                  // 32768 f32 scratch (128 KB)

  nns_norms_kernel<<<Q_MAX / 8, 256, 0, stream>>>(data, queue, norms);
  nns_knn_kernel<<<B_ROWS / 16, 256, 0, stream>>>(data, queue, jdx, norms, out);
}